// Features_75290776698888
// MI455X (gfx1250) — compile-verified
//
#include <hip/hip_runtime.h>
#include <math.h>

#define RES_CFG   4096
#define NUM_FREQ  8

typedef float v4f __attribute__((ext_vector_type(4)));

__device__ __forceinline__ int wrapi(int v, int m) {
    int r = v % m;
    return (r < 0) ? r + m : r;   // Python-style modulo (v can be -1)
}

__device__ __forceinline__ float triw(float t) {
    float f = t - floorf(t);
    return 1.0f - 4.0f * fabsf(f - 0.5f);
}

// C0T/C1T > 0: compile-time channel counts (fully unrolled gather loops).
// C0T/C1T == 0: generic runtime fallback.
template <int C0T, int C1T>
__global__ __launch_bounds__(256) void features_kernel(
    const int*   __restrict__ cs,     // (B,2) int32 coordinate_start
    const float* __restrict__ grid0,  // (C0,H0,W0)
    const float* __restrict__ grid1,  // (C1,H1,W1)
    const int*   __restrict__ h_ptr,
    const int*   __restrict__ w_ptr,
    const int*   __restrict__ lod_ptr,
    float*       __restrict__ out,    // (B, C0+C1+1+2F, h, w)
    int B, int C0r, int H0, int W0, int C1r, int H1, int W1)
{
    const int C0 = (C0T > 0) ? C0T : C0r;
    const int C1 = (C1T > 0) ? C1T : C1r;

    const int h   = *h_ptr;
    const int w   = *w_ptr;
    const int lod = *lod_ptr;
    const int res = RES_CFG >> lod;

    const int C  = C0 + C1 + 1 + 2 * NUM_FREQ;
    const int wq = (w + 3) >> 2;                 // 4-pixel groups per row
    const int rowgrp = h * wq;
    const int total  = B * rowgrp;               // fits int (out_size is int)
    const int plane  = h * w;

    const float sy1     = (float)H1 / (float)res;
    const float sx1     = (float)W1 / (float)res;
    const float inv_res = 1.0f / (float)res;

    const float lod_levels = log2f((float)RES_CFG);    // 12
    const float lod_offset = lod_levels * 0.5f;        // 6
    const float lod_scale  = 2.0f / lod_offset;        // 1/3
    const float lod_val    = ((float)lod - lod_offset) / lod_scale;

    const bool vec4 = ((w & 3) == 0);            // full, 16B-aligned groups

    for (int g = blockIdx.x * blockDim.x + threadIdx.x;
         g < total;
         g += gridDim.x * blockDim.x) {

        const int b  = g / rowgrp;
        const int r  = g - b * rowgrp;
        const int y  = r / wq;
        const int xq = (r - y * wq) << 2;

        const int ys  = cs[2 * b]     + y;       // >= 0 always
        const int xs0 = cs[2 * b + 1] + xq;

        // ---------- per-row (shared by the 4 sub-pixels) ----------
        const int   iy0 = ((ys * H0) / res) % H0;      // nearest row, grid0
        const float fy  = (float)ys * sy1 - 0.5f;
        const float y0f = floorf(fy);
        const float ty  = fy - y0f;
        const int   y0  = wrapi((int)y0f, H1);
        const int   y1  = (y0 + 1 == H1) ? 0 : y0 + 1;

        float pyv[NUM_FREQ];
        {
            const float uy = (float)ys * inv_res;
            float fr = 1.0f;
            #pragma unroll
            for (int k = 0; k < NUM_FREQ; ++k) { pyv[k] = triw(uy * fr); fr *= 2.0f; }
        }

        // ---------- per sub-pixel ----------
        int   ix0[4], x0i[4], x1i[4];
        float tx[4], pxv[NUM_FREQ][4];
        #pragma unroll
        for (int i = 0; i < 4; ++i) {
            const int xg = xs0 + i;
            ix0[i] = ((xg * W0) / res) % W0;
            const float fx  = (float)xg * sx1 - 0.5f;
            const float x0f = floorf(fx);
            tx[i] = fx - x0f;
            const int x0 = wrapi((int)x0f, W1);
            x0i[i] = x0;
            x1i[i] = (x0 + 1 == W1) ? 0 : x0 + 1;
            const float ux = (float)xg * inv_res;
            float fr = 1.0f;
            #pragma unroll
            for (int k = 0; k < NUM_FREQ; ++k) { pxv[k][i] = triw(ux * fr); fr *= 2.0f; }
        }

        // gfx1250 prefetch (global_prefetch_b8) on the rows we are about to gather
        __builtin_prefetch(grid0 + iy0 * W0 + ix0[0], 0, 3);
        __builtin_prefetch(grid1 + y0  * W1 + x0i[0], 0, 3);
        __builtin_prefetch(grid1 + y1  * W1 + x0i[0], 0, 3);

        float* outp = out + b * C * plane + y * w + xq;

        // Compute all channels into registers, then one straight store run.
        auto compute_store = [&](auto storer) {
            // ---------- grid0: nearest-texel gather ----------
            #pragma unroll
            for (int c = 0; c < C0; ++c) {
                const float* g0p = grid0 + (c * H0 + iy0) * W0;
                v4f v;
                v[0] = g0p[ix0[0]]; v[1] = g0p[ix0[1]];
                v[2] = g0p[ix0[2]]; v[3] = g0p[ix0[3]];
                storer(c, v);
            }
            // ---------- grid1: bilinear gather ----------
            #pragma unroll
            for (int c = 0; c < C1; ++c) {
                const float* r0 = grid1 + (c * H1 + y0) * W1;
                const float* r1 = grid1 + (c * H1 + y1) * W1;
                v4f v;
                #pragma unroll
                for (int i = 0; i < 4; ++i) {
                    const float c00 = r0[x0i[i]], c01 = r0[x1i[i]];
                    const float c10 = r1[x0i[i]], c11 = r1[x1i[i]];
                    const float top = c00 + tx[i] * (c01 - c00);
                    const float bot = c10 + tx[i] * (c11 - c10);
                    v[i] = top + ty * (bot - top);
                }
                storer(C0 + c, v);
            }
            // ---------- LOD channel ----------
            {
                v4f v = { lod_val, lod_val, lod_val, lod_val };
                storer(C0 + C1, v);
            }
            // ---------- positional encoding: py then px ----------
            #pragma unroll
            for (int k = 0; k < NUM_FREQ; ++k) {
                v4f v = { pyv[k], pyv[k], pyv[k], pyv[k] };
                storer(C0 + C1 + 1 + k, v);
            }
            #pragma unroll
            for (int k = 0; k < NUM_FREQ; ++k) {
                v4f v = { pxv[k][0], pxv[k][1], pxv[k][2], pxv[k][3] };
                storer(C0 + C1 + 1 + NUM_FREQ + k, v);
            }
        };

        if (vec4) {
            // hot path: 33 NT b128 stores, no per-store branching
            compute_store([&](int c, v4f v) {
                __builtin_nontemporal_store(v, (v4f*)(outp + c * plane));
            });
        } else {
            const int rem = w - xq;
            compute_store([&](int c, v4f v) {
                float* p = outp + c * plane;
                if (rem > 0) p[0] = v[0];
                if (rem > 1) p[1] = v[1];
                if (rem > 2) p[2] = v[2];
                if (rem > 3) p[3] = v[3];
            });
        }
    }
}

extern "C" void kernel_launch(void* const* d_in, const int* in_sizes, int n_in,
                              void* d_out, int out_size, void* d_ws, size_t ws_size,
                              hipStream_t stream) {
    const int*   cs    = (const int*)  d_in[0];
    const float* grid0 = (const float*)d_in[1];
    const float* grid1 = (const float*)d_in[2];
    const int*   hp    = (const int*)  d_in[3];
    const int*   wp    = (const int*)  d_in[4];
    const int*   lodp  = (const int*)  d_in[5];
    float*       out   = (float*)      d_out;

    const int B = in_sizes[0] / 2;

    // Grids are square (C, H, H) with C = 8 per the reference config; derive H.
    auto derive = [](long long n, int C, int& H) {
        long long hw = n / C;
        int hh = (int)llround(sqrt((double)hw));
        while ((long long)hh * hh > hw) --hh;
        while ((long long)(hh + 1) * (hh + 1) <= hw) ++hh;
        H = hh;
    };
    int C0 = 8, H0, C1 = 8, H1;
    derive(in_sizes[1], C0, H0);   // 1024
    derive(in_sizes[2], C1, H1);   // 512

    const int C = C0 + C1 + 1 + 2 * NUM_FREQ;          // 33
    const long long pixels = (long long)out_size / C;  // B*h*w
    long long groups = (pixels + 3) / 4;               // 4-wide x groups (lower bound)
    int blocks = (int)((groups + 255) / 256);
    if (blocks < 1) blocks = 1;
    // grid-stride loop in the kernel covers any rounding (w % 4 != 0) cases

    if (C0 == 8 && C1 == 8) {
        features_kernel<8, 8><<<blocks, 256, 0, stream>>>(
            cs, grid0, grid1, hp, wp, lodp, out, B, C0, H0, H0, C1, H1, H1);
    } else {
        features_kernel<0, 0><<<blocks, 256, 0, stream>>>(
            cs, grid0, grid1, hp, wp, lodp, out, B, C0, H0, H0, C1, H1, H1);
    }
}